// RadiusFilter_27238682591584
// MI455X (gfx1250) — compile-verified
//
#include <hip/hip_runtime.h>
#include <hip/hip_bf16.h>

typedef __attribute__((ext_vector_type(2))) float v2f;
typedef __attribute__((ext_vector_type(8))) float v8f;

#define BB 8
#define LL 4096
#define R2 0.01f
#define NNEIGH 8

// One wave = one 16-wide i-tile. Loop over 256 j-tiles; each tile is one
// V_WMMA_F32_16X16X4_F32 producing the 16x16 squared-distance block directly:
//   A row j = (x_j, y_j, |r_j|^2, 1)
//   B col i = (-2 x_i, -2 y_i, 1, |r_i|^2)
//   D[j][i] = |r_j|^2 + |r_i|^2 - 2 r_j.r_i = d^2(i,j)
__global__ __launch_bounds__(256) void radius_count_kernel(
    const float* __restrict__ x, const float* __restrict__ y,
    const unsigned char* __restrict__ valid, float* __restrict__ out_mask) {
  const int lane = threadIdx.x & 31;
  const int wave = threadIdx.x >> 5;
  const int batch = blockIdx.x >> 5;       // 32 blocks per batch
  const int blockRow = blockIdx.x & 31;    // 8 i-tiles per block
  const int iTile = blockRow * 8 + wave;
  const int i0 = iTile * 16;

  const float* xb = x + batch * LL;
  const float* yb = y + batch * LL;

  const int li = lane & 15;
  const bool hiHalf = lane >= 16;

  // B-matrix (fixed per wave): 4x16, VGPR0 = {K0 | K2}, VGPR1 = {K1 | K3}
  const float ix = xb[i0 + li];
  const float iy = yb[i0 + li];
  const float ni = ix * ix + iy * iy;
  v2f bmat;
  bmat.x = hiHalf ? 1.0f : (-2.0f * ix);   // K0 = -2x_i ; K2 = 1
  bmat.y = hiHalf ? ni : (-2.0f * iy);     // K1 = -2y_i ; K3 = |r_i|^2

  int cnt = 0;

#pragma unroll 4
  for (int jt = 0; jt < LL / 16; ++jt) {
    const int j0 = jt * 16;
    const float jx = xb[j0 + li];
    const float jy = yb[j0 + li];
    const float nj = jx * jx + jy * jy;

    // A-matrix: 16x4, lanes 0-15 hold K0,K1; lanes 16-31 hold K2,K3
    v2f amat;
    amat.x = hiHalf ? nj : jx;             // K0 = x_j ; K2 = |r_j|^2
    amat.y = hiHalf ? 1.0f : jy;           // K1 = y_j ; K3 = 1

    v8f c = {};
    c = __builtin_amdgcn_wmma_f32_16x16x4_f32(
        /*neg_a=*/false, amat, /*neg_b=*/false, bmat,
        /*c_mod=*/(short)0, c, /*reuse_a=*/false, /*reuse_b=*/false);

    // C/D layout: lane holds column N = lane%16, rows r + 8*(lane/16).
#pragma unroll
    for (int r = 0; r < 8; ++r) cnt += (c[r] <= R2) ? 1 : 0;
  }

  // Column i lives in lanes i (rows 0-7) and i+16 (rows 8-15): fold halves.
  cnt += __shfl_xor(cnt, 16, 32);

  if (lane < 16) {
    const int i = i0 + lane;
    const bool keep = cnt > NNEIGH;
    const bool v = valid[batch * LL + i] != 0;
    out_mask[batch * LL + i] = (keep && v) ? 1.0f : 0.0f;
  }
}

// Pack p, y, x, t (copies) and target (int64 -> float) into the concatenated
// output. Mask slot (index 4) is written by radius_count_kernel.
__global__ void pack_outputs_kernel(const float* __restrict__ p,
                                    const float* __restrict__ y,
                                    const float* __restrict__ x,
                                    const float* __restrict__ t,
                                    const long long* __restrict__ target,
                                    float* __restrict__ out) {
  const int n = BB * LL;
  const int gid = blockIdx.x * blockDim.x + threadIdx.x;
  if (gid < n) {
    out[gid] = p[gid];
  } else if (gid < 2 * n) {
    out[gid] = y[gid - n];
  } else if (gid < 3 * n) {
    out[gid] = x[gid - 2 * n];
  } else if (gid < 4 * n) {
    out[gid] = t[gid - 3 * n];
  } else if (gid < 4 * n + BB) {
    const int b = gid - 4 * n;
    out[5 * n + b] = (float)target[b];
  }
}

extern "C" void kernel_launch(void* const* d_in, const int* in_sizes, int n_in,
                              void* d_out, int out_size, void* d_ws, size_t ws_size,
                              hipStream_t stream) {
  const float* p = (const float*)d_in[0];
  const float* y = (const float*)d_in[1];
  const float* x = (const float*)d_in[2];
  const float* t = (const float*)d_in[3];
  const unsigned char* valid = (const unsigned char*)d_in[4];
  const long long* target = (const long long*)d_in[5];

  float* out = (float*)d_out;
  float* out_mask = out + 4 * BB * LL;

  // 8 batches * 32 blocks, 8 waves per block, one 16-row i-tile per wave.
  radius_count_kernel<<<BB * 32, 256, 0, stream>>>(x, y, valid, out_mask);

  const int total = 4 * BB * LL + BB;
  pack_outputs_kernel<<<(total + 255) / 256, 256, 0, stream>>>(p, y, x, t,
                                                               target, out);
}